// TopKTokenSelector_23811298689624
// MI455X (gfx1250) — compile-verified
//
#include <hip/hip_runtime.h>
#include <math.h>

typedef float v2f __attribute__((ext_vector_type(2)));
typedef float v8f __attribute__((ext_vector_type(8)));

#define B_ 32
#define L_ 4096
#define D_ 768
#define K_ 1024
#define U_ 3072   // L - K

// ---------------------------------------------------------------------------
// Kernel 1: score_logits[b*L + l] = dot(inputs[b,l,:], w)  via v_wmma_f32_16x16x4_f32
// block = 128 threads = 4 waves; each wave computes 16 token scores.
// ---------------------------------------------------------------------------
__global__ __launch_bounds__(128) void score_kernel(const float* __restrict__ in,
                                                    const float* __restrict__ w,
                                                    float* __restrict__ scores) {
    __shared__ float wsh[D_];
    __shared__ float tile[4][16][68];   // pad 68: A-reads hit distinct banks

    const int tid = threadIdx.x;
    for (int i = tid; i < D_; i += 128) wsh[i] = w[i];
    __syncthreads();

    const int wave = tid >> 5;
    const int lane = tid & 31;
    const int m    = lane & 15;   // A-matrix row M
    const int kh   = lane >> 4;   // 0: K=0,1   1: K=2,3
    const long long tok0 = (long long)blockIdx.x * 64 + wave * 16;
    const float* src = in + tok0 * D_;

    v8f acc = {};

    for (int k0 = 0; k0 < D_; k0 += 64) {
        // Stage 16x64 chunk into wave-private LDS, coalesced (16 lanes x float4 per row-pair)
#pragma unroll
        for (int j = 0; j < 8; ++j) {
            int t   = j * 32 + lane;        // 0..255 float4 slots
            int row = t >> 4;
            int c4  = t & 15;
            const float4 v = *(const float4*)(src + row * D_ + k0 + c4 * 4);
            float* dst = &tile[wave][row][c4 * 4];
            dst[0] = v.x; dst[1] = v.y; dst[2] = v.z; dst[3] = v.w;
        }
        // same-wave LDS ops are in-order: store->load RAW is safe without barrier
#pragma unroll
        for (int kk = 0; kk < 64; kk += 4) {
            v2f a;
            a.x = tile[wave][m][kk + 2 * kh];
            a.y = tile[wave][m][kk + 2 * kh + 1];
            // Degenerate B (4x16): only column N=0 nonzero = w chunk.
            // lane 0 holds B[0][0],B[1][0]; lane 16 holds B[2][0],B[3][0].
            float wa = wsh[k0 + kk + 2 * kh];
            float wb = wsh[k0 + kk + 2 * kh + 1];
            v2f bfrag;
            bfrag.x = (m == 0) ? wa : 0.0f;   // branch-free: EXEC stays all-1s
            bfrag.y = (m == 0) ? wb : 0.0f;
            acc = __builtin_amdgcn_wmma_f32_16x16x4_f32(
                false, a, false, bfrag, (short)0, acc, false, false);
        }
    }
    // D column N=0: lane 0 -> M=0..7 in acc[0..7]; lane 16 -> M=8..15
    if (lane == 0) {
#pragma unroll
        for (int j = 0; j < 8; ++j) scores[tok0 + j] = acc[j];
    } else if (lane == 16) {
#pragma unroll
        for (int j = 0; j < 8; ++j) scores[tok0 + 8 + j] = acc[j];
    }
}

// ---------------------------------------------------------------------------
// Kernel 2: per batch: full bitonic sort of 4096 keys (desc score, asc index),
// selected softmax weights, and softmax weights over scores[0:3072].
// ---------------------------------------------------------------------------
__global__ __launch_bounds__(1024) void topk_softmax_kernel(const float* __restrict__ scores,
                                                            int* __restrict__ selidx,
                                                            float* __restrict__ selw,
                                                            float* __restrict__ smu) {
    const int b   = blockIdx.x;
    const int tid = threadIdx.x;
    __shared__ unsigned long long keys[L_];
    __shared__ float red[1024];

    const float* sc = scores + (long long)b * L_;

    for (int i = tid; i < L_; i += 1024) {
        unsigned u = __float_as_uint(sc[i]);
        u = (u & 0x80000000u) ? ~u : (u | 0x80000000u);       // ascending-orderable
        keys[i] = ((unsigned long long)(~u) << 32) | (unsigned)i; // asc sort => desc score, asc idx
    }
    __syncthreads();

    for (int k = 2; k <= L_; k <<= 1) {
        for (int j = k >> 1; j > 0; j >>= 1) {
            for (int i = tid; i < L_; i += 1024) {
                int p = i ^ j;
                if (p > i) {
                    bool asc = ((i & k) == 0);
                    unsigned long long x = keys[i], y = keys[p];
                    if ((x > y) == asc) { keys[i] = y; keys[p] = x; }
                }
            }
            __syncthreads();
        }
    }

    // ---- selected softmax (positions 0..1023, sorted descending) ----
    int   idx  = (int)(keys[tid] & 0xFFFFFFFFull);
    int   idx0 = (int)(keys[0]   & 0xFFFFFFFFull);
    float sv   = sc[idx];
    float smax = sc[idx0];
    float e    = expf(sv - smax);
    red[tid] = e;
    __syncthreads();
    for (int s = 512; s > 0; s >>= 1) {
        if (tid < s) red[tid] += red[tid + s];
        __syncthreads();
    }
    float denom = red[0];
    selidx[b * K_ + tid] = idx;
    selw  [b * K_ + tid] = e / denom;
    __syncthreads();

    // ---- unselected softmax over scores[b, 0:3072] (mask quirk => indices 0..3071) ----
    float mx = -__builtin_inff();
    for (int i = tid; i < U_; i += 1024) mx = fmaxf(mx, sc[i]);
    red[tid] = mx;
    __syncthreads();
    for (int s = 512; s > 0; s >>= 1) {
        if (tid < s) red[tid] = fmaxf(red[tid], red[tid + s]);
        __syncthreads();
    }
    mx = red[0];
    __syncthreads();
    float sum = 0.0f;
    for (int i = tid; i < U_; i += 1024) sum += expf(sc[i] - mx);
    red[tid] = sum;
    __syncthreads();
    for (int s = 512; s > 0; s >>= 1) {
        if (tid < s) red[tid] += red[tid + s];
        __syncthreads();
    }
    float d3 = red[0];
    for (int i = tid; i < U_; i += 1024)
        smu[b * U_ + i] = expf(sc[i] - mx) / d3;
}

// ---------------------------------------------------------------------------
// Kernel 3: out[b, i, :] = inputs[b, selidx[b,i], :] * selw[b,i]   (i < 1024)
// one block (192 threads, float4) per output row
// ---------------------------------------------------------------------------
__global__ __launch_bounds__(192) void gather_kernel(const float* __restrict__ in,
                                                     const int* __restrict__ selidx,
                                                     const float* __restrict__ selw,
                                                     float* __restrict__ out) {
    const int b = blockIdx.y, i = blockIdx.x, t = threadIdx.x;
    const int   idx = selidx[b * K_ + i];
    const float wgt = selw [b * K_ + i];
    const float4* src = (const float4*)(in  + ((long long)b * L_ + idx) * D_);
    float4*       dst = (float4*)      (out + ((long long)b * (K_ + 1) + i) * D_);
    float4 v = src[t];
    v.x *= wgt; v.y *= wgt; v.z *= wgt; v.w *= wgt;
    dst[t] = v;
}

// ---------------------------------------------------------------------------
// Kernel 4: partial sums of Σ_l smu[b,l] * inputs[b,l,:] over l-slices of 384
// ---------------------------------------------------------------------------
__global__ __launch_bounds__(192) void unsel_partial_kernel(const float* __restrict__ in,
                                                            const float* __restrict__ smu,
                                                            float* __restrict__ part) {
    const int b = blockIdx.y, g = blockIdx.x, t = threadIdx.x;
    const float4* src = (const float4*)(in + (long long)b * L_ * D_);
    float4 acc = {0.f, 0.f, 0.f, 0.f};
    const int l0 = g * 384;
    for (int l = l0; l < l0 + 384; ++l) {
        float wv = smu[b * U_ + l];
        float4 v = src[(long long)l * (D_ / 4) + t];
        acc.x += wv * v.x; acc.y += wv * v.y; acc.z += wv * v.z; acc.w += wv * v.w;
    }
    ((float4*)part)[((long long)b * 8 + g) * (D_ / 4) + t] = acc;
}

// ---------------------------------------------------------------------------
// Kernel 5: reduce 8 partials -> out[b, 1024, :]
// ---------------------------------------------------------------------------
__global__ __launch_bounds__(192) void unsel_final_kernel(const float* __restrict__ part,
                                                          float* __restrict__ out) {
    const int b = blockIdx.x, t = threadIdx.x;
    float4 acc = {0.f, 0.f, 0.f, 0.f};
    for (int g = 0; g < 8; ++g) {
        float4 v = ((const float4*)part)[((long long)b * 8 + g) * (D_ / 4) + t];
        acc.x += v.x; acc.y += v.y; acc.z += v.z; acc.w += v.w;
    }
    ((float4*)out)[((long long)b * (K_ + 1) + K_) * (D_ / 4) + t] = acc;
}

// ---------------------------------------------------------------------------
extern "C" void kernel_launch(void* const* d_in, const int* in_sizes, int n_in,
                              void* d_out, int out_size, void* d_ws, size_t ws_size,
                              hipStream_t stream) {
    const float* in = (const float*)d_in[0];   // [32, 4096, 768] f32
    const float* w  = (const float*)d_in[1];   // [768, 1] f32
    float* out = (float*)d_out;                // [32, 1025, 768] f32

    // workspace layout (floats): scores | selidx | selw | smu | partials
    float* scores = (float*)d_ws;              // 32*4096      = 131072
    int*   selidx = (int*)(scores + B_ * L_);  // 32*1024      =  32768
    float* selw   = (float*)(selidx + B_ * K_);// 32*1024      =  32768
    float* smu    = selw + B_ * K_;            // 32*3072      =  98304
    float* part   = smu + B_ * U_;             // 32*8*768     = 196608 (byte offset 1179648, 16B aligned)

    score_kernel       <<<(B_ * L_) / 64, 128, 0, stream>>>(in, w, scores);
    topk_softmax_kernel<<<B_, 1024, 0, stream>>>(scores, selidx, selw, smu);
    gather_kernel      <<<dim3(K_, B_), 192, 0, stream>>>(in, selidx, selw, out);
    unsel_partial_kernel<<<dim3(8, B_), 192, 0, stream>>>(in, smu, part);
    unsel_final_kernel <<<B_, 192, 0, stream>>>(part, out);
}